// QLstm_18622978195482
// MI455X (gfx1250) — compile-verified
//
#include <hip/hip_runtime.h>
#include <math.h>

typedef float v2f __attribute__((ext_vector_type(2)));
typedef float v4f __attribute__((ext_vector_type(4)));
typedef float v8f __attribute__((ext_vector_type(8)));

#define NW 6            // wires
#define NA 64           // amplitudes (2^6)
#define TT 256          // time steps
#define FD 8            // feature dim
#define TB 16           // samples per block tile (WMMA N)
#define FRAG 8192       // floats per circuit in fragment layout: 4mt*32lane*16kk*4

// true global (addrspace 1) vector load: emits global_load_b128, not flat
typedef __attribute__((address_space(1))) const v4f* gcv4p;
__device__ __forceinline__ v4f load_g4(const float* p) {
  return *(gcv4p)(unsigned long long)p;
}

// ---------------------------------------------------------------------------
// Kernel 1: build the six 64x64 complex unitaries U_c from phi (6,2,18) and
// store them pre-swizzled in WMMA A-fragment order:
//   Uf[c][mt][lane][kk][4] = { Ur[m][k0], Ur[m][k0+1], Ui[m][k0], Ui[m][k0+1] }
//   with m = mt*16 + (lane&15), k0 = kk*4 + ((lane>>4)<<1)
// ---------------------------------------------------------------------------
__device__ __noinline__ void apply_gate_col(float* re, float* im, int col, int wire,
    float g00r, float g00i, float g01r, float g01i,
    float g10r, float g10i, float g11r, float g11i) {
  const int b = 5 - wire;            // wire 0 = most significant bit
  const int low = (1 << b) - 1;
  for (int p = 0; p < 32; ++p) {
    const int i0 = ((p & ~low) << 1) | (p & low);
    const int i1 = i0 | (1 << b);
    const float a0r = re[i0*NA+col], a0i = im[i0*NA+col];
    const float a1r = re[i1*NA+col], a1i = im[i1*NA+col];
    re[i0*NA+col] = g00r*a0r - g00i*a0i + g01r*a1r - g01i*a1i;
    im[i0*NA+col] = g00r*a0i + g00i*a0r + g01r*a1i + g01i*a1r;
    re[i1*NA+col] = g10r*a0r - g10i*a0i + g11r*a1r - g11i*a1i;
    im[i1*NA+col] = g10r*a0i + g10i*a0r + g11r*a1i + g11i*a1r;
  }
}

__device__ __noinline__ void cnot_col(float* re, float* im, int col, int ctrl, int tgt) {
  const int bc_ = 5 - ctrl, bt = 5 - tgt;
  for (int a = 0; a < NA; ++a) {
    if (((a >> bc_) & 1) && !((a >> bt) & 1)) {
      const int a2 = a | (1 << bt);
      float tr = re[a*NA+col]; re[a*NA+col] = re[a2*NA+col]; re[a2*NA+col] = tr;
      float ti = im[a*NA+col]; im[a*NA+col] = im[a2*NA+col]; im[a2*NA+col] = ti;
    }
  }
}

__global__ void __launch_bounds__(64) build_unitaries(
    const float* __restrict__ phi, float* __restrict__ Uf) {
  __shared__ float re[NA*NA];
  __shared__ float im[NA*NA];
  const int col = threadIdx.x;       // 0..63, owns column col
  const int cir = blockIdx.x;        // 0..5
  for (int a = 0; a < NA; ++a) { re[a*NA+col] = (a == col) ? 1.f : 0.f; im[a*NA+col] = 0.f; }
  const float* w = phi + cir * 2 * 18;
  for (int l = 0; l < 2; ++l) {
    for (int i = 0; i < NW; ++i) cnot_col(re, im, col, i, (i + 1) % NW);
    for (int i = 0; i < NW; ++i) cnot_col(re, im, col, i, (i + 2) % NW);
    for (int i = 0; i < NW; ++i) {
      const float tx = 0.5f * w[l*18 + 3*i + 0];   // RX
      apply_gate_col(re, im, col, i, cosf(tx), 0.f, 0.f, -sinf(tx),
                                     0.f, -sinf(tx), cosf(tx), 0.f);
      const float ty = 0.5f * w[l*18 + 3*i + 1];   // RY
      apply_gate_col(re, im, col, i, cosf(ty), 0.f, -sinf(ty), 0.f,
                                     sinf(ty), 0.f, cosf(ty), 0.f);
      const float tz = 0.5f * w[l*18 + 3*i + 2];   // RZ
      apply_gate_col(re, im, col, i, cosf(tz), -sinf(tz), 0.f, 0.f,
                                     0.f, 0.f, cosf(tz), sinf(tz));
    }
  }
  __syncthreads();
  // emit fragment-swizzled layout
  for (int idx = col; idx < 4*32*16; idx += 64) {
    const int kk = idx & 15;
    const int ln = (idx >> 4) & 31;
    const int mt = idx >> 9;
    const int m  = mt*16 + (ln & 15);
    const int k0 = kk*4 + ((ln >> 4) << 1);
    float* dst = Uf + (size_t)cir*FRAG + (size_t)idx*4;
    dst[0] = re[m*NA + k0];
    dst[1] = re[m*NA + k0 + 1];
    dst[2] = im[m*NA + k0];
    dst[3] = im[m*NA + k0 + 1];
  }
}

// ---------------------------------------------------------------------------
// Kernel 2: QLSTM loop. 4 waves/block, 16 samples/block; per step the four
// gate circuits run on separate waves (independent GEMMs), results meet in LDS.
// circuit eval = complex (64x64)x(64x16) GEMM via V_WMMA_F32_16X16X4_F32.
// ---------------------------------------------------------------------------
__device__ __forceinline__ float sigm(float v) { return 1.f / (1.f + expf(-v)); }

__device__ __forceinline__ v8f wmma4(v2f a, v2f b, v8f c) {
  return __builtin_amdgcn_wmma_f32_16x16x4_f32(false, a, false, b, (short)0, c,
                                               false, false);
}

struct Ev6 { float v[NW]; };

__device__ __noinline__ Ev6 circuit_ev(const float* __restrict__ uf,
                                       const float* psiR, const float* psiI,
                                       int lane) {
  const int n     = lane & 15;            // sample / N index (dup in hi half)
  const int khalf = (lane >> 4) << 1;     // lanes 16-31 hold K=+2,+3
  const int mhi   = (lane >> 4) << 3;     // D rows: +8 for hi half
  const float* prow  = psiR + n * NA;
  const float* pirow = psiI + n * NA;
  float acc[NW] = {0.f, 0.f, 0.f, 0.f, 0.f, 0.f};
  for (int mt = 0; mt < 4; ++mt) {        // 64 output rows = 4 M-tiles of 16
    v8f dr = {0.f,0.f,0.f,0.f,0.f,0.f,0.f,0.f};
    v8f di = {0.f,0.f,0.f,0.f,0.f,0.f,0.f,0.f};
    const float* fb = uf + ((mt*32 + lane) << 6);   // 64 floats per (mt,lane)
    __builtin_prefetch(fb + (32 << 6), 0, 1);       // pull next M-tile block
    #pragma unroll 2
    for (int kk = 0; kk < 16; ++kk) {     // K = 64 in chunks of 4
      const v4f u = load_g4(fb + kk*4);   // {Ur0,Ur1,Ui0,Ui1} one b128 load
      const int k0 = kk*4 + khalf;
      v2f Prf = *(const v2f*)(prow + k0);
      v2f Pif = *(const v2f*)(pirow + k0);
      v2f Urf;  Urf.x  = u.x;  Urf.y  = u.y;
      v2f Uif;  Uif.x  = u.z;  Uif.y  = u.w;
      v2f nUif; nUif.x = -u.z; nUif.y = -u.w;
      dr = wmma4(Urf,  Prf, dr);          // Re += Ur*Pr
      dr = wmma4(nUif, Pif, dr);          // Re -= Ui*Pi
      di = wmma4(Urf,  Pif, di);          // Im += Ur*Pi
      di = wmma4(Uif,  Prf, di);          // Im += Ui*Pr
    }
    #pragma unroll
    for (int r = 0; r < 8; ++r) {         // D row M = mt*16 + mhi + r
      const float p = dr[r]*dr[r] + di[r]*di[r];
      const int m = mt*16 + mhi + r;
      #pragma unroll
      for (int i = 0; i < NW; ++i)
        acc[i] += ((m >> (5 - i)) & 1) ? -p : p;   // <Z_i> = P(0)-P(1)
    }
  }
  Ev6 e;
  #pragma unroll
  for (int i = 0; i < NW; ++i)            // fold the two lane halves
    e.v[i] = acc[i] + __shfl_xor(acc[i], 16, 32);
  return e;
}

// product-state embedding: psi = prod_i RZ(atan(x^2)) RY(atan(x)) |0>
// each wave writes amplitude range [a0,a1) for its 16 samples
__device__ __forceinline__ void build_psi_part(const float* ang, float* psiR,
                                               float* psiI, int s, bool writer,
                                               int a0, int a1) {
  float r0[NW], i0a[NW], r1[NW], i1a[NW];
  #pragma unroll
  for (int i = 0; i < NW; ++i) {
    const float e1 = 0.5f * atanf(ang[i]);
    const float e2 = 0.5f * atanf(ang[i] * ang[i]);
    const float c1 = cosf(e1), s1 = sinf(e1);
    const float c2 = cosf(e2), s2 = sinf(e2);
    r0[i] = c1 * c2;  i0a[i] = -c1 * s2;   // e^{-i e2/2} cos(e1/2)
    r1[i] = s1 * c2;  i1a[i] =  s1 * s2;   // e^{+i e2/2} sin(e1/2)
  }
  if (writer) {
    for (int a = a0; a < a1; ++a) {
      float re = 1.f, im = 0.f;
      #pragma unroll
      for (int i = 0; i < NW; ++i) {
        const int bit = (a >> (5 - i)) & 1;
        const float fr = bit ? r1[i] : r0[i];
        const float fi = bit ? i1a[i] : i0a[i];
        const float nr = re * fr - im * fi;
        im = re * fi + im * fr;
        re = nr;
      }
      psiR[s*NA + a] = re;
      psiI[s*NA + a] = im;
    }
  }
}

__global__ void __launch_bounds__(128) qlstm_main(
    const float* __restrict__ x,  const float* __restrict__ Wc,
    const float* __restrict__ bc, const float* __restrict__ Uf,
    float* __restrict__ out, int B) {
  __shared__ float psiR[TB*NA];           // B-matrix: [sample][amp], re
  __shared__ float psiI[TB*NA];           // im
  __shared__ float sEv[6*TB*NW];          // expvals per circuit/sample/wire
  __shared__ float sWc[NW*(FD+NW)];
  __shared__ float sBc[NW];
  const int tid  = threadIdx.x;
  const int wave = tid >> 5;              // 0..3
  const int lane = tid & 31;
  const int s = lane & 15;                // sample within tile
  const int sample = blockIdx.x * TB + s;
  const bool wr = lane < TB;
  for (int idx = tid; idx < NW*(FD+NW); idx += 128) sWc[idx] = Wc[idx];
  if (tid < NW) sBc[tid] = bc[tid];
  __syncthreads();

  float cst[NW], hst[NW];
  #pragma unroll
  for (int i = 0; i < NW; ++i) { cst[i] = 0.f; hst[i] = 0.f; }

  for (int t = 0; t < TT; ++t) {
    // feature combiner: X = 2*sigmoid([x_t, h] @ Wc^T + bc) - 1  (dup per wave)
    float z[FD+NW];
    #pragma unroll
    for (int j = 0; j < FD; ++j) z[j] = x[((size_t)sample*TT + t)*FD + j];
    #pragma unroll
    for (int j = 0; j < NW; ++j) z[FD+j] = hst[j];
    float X[NW];
    #pragma unroll
    for (int i = 0; i < NW; ++i) {
      float a = sBc[i];
      #pragma unroll
      for (int j = 0; j < FD+NW; ++j) a += sWc[i*(FD+NW)+j] * z[j];
      X[i] = 2.f * sigm(a) - 1.f;
    }
    build_psi_part(X, psiR, psiI, s, wr, wave*16, wave*16 + 16);
    __syncthreads();                                 // (A) psi(X) ready

    // phase 1: forget/input/cell/output circuits, one per wave
    {
      Ev6 e = circuit_ev(Uf + (size_t)wave*FRAG, psiR, psiI, lane);
      if (wr) {
        #pragma unroll
        for (int i = 0; i < NW; ++i) sEv[(wave*TB + s)*NW + i] = e.v[i];
      }
    }
    __syncthreads();                                 // (B) ev[0..3] ready

    float resc[NW];
    #pragma unroll
    for (int i = 0; i < NW; ++i) {
      const float ft = sigm(sEv[(0*TB + s)*NW + i]);
      const float it = sigm(sEv[(1*TB + s)*NW + i]);
      const float Ct = sigm(sEv[(2*TB + s)*NW + i]);
      const float ot = sigm(sEv[(3*TB + s)*NW + i]);
      cst[i]  = ft * cst[i] + it * Ct;
      resc[i] = ot * tanhf(cst[i]);
    }
    build_psi_part(resc, psiR, psiI, s, wr, wave*16, wave*16 + 16);
    __syncthreads();                                 // (C) psi(resc) ready

    // phase 2: h and y circuits on waves 0 and 1
    if (wave < 2) {
      Ev6 e = circuit_ev(Uf + (size_t)(4 + wave)*FRAG, psiR, psiI, lane);
      if (wr) {
        #pragma unroll
        for (int i = 0; i < NW; ++i) sEv[((4 + wave)*TB + s)*NW + i] = e.v[i];
      }
    }
    __syncthreads();                                 // (D) ev[4..5] ready

    #pragma unroll
    for (int i = 0; i < NW; ++i) hst[i] = sEv[(4*TB + s)*NW + i];
    if (wave == 2 && wr) {
      #pragma unroll
      for (int i = 0; i < NW; ++i)
        out[((size_t)sample*TT + t)*NW + i] = sEv[(5*TB + s)*NW + i];
    }
  }
  if (wave == 0 && wr) {
    const size_t base = (size_t)B * TT * NW;
    #pragma unroll
    for (int i = 0; i < NW; ++i) {
      out[base + (size_t)sample*NW + i] = cst[i];                    // final c
      out[base + (size_t)B*NW + (size_t)sample*NW + i] = hst[i];     // final h
    }
  }
}

// ---------------------------------------------------------------------------
extern "C" void kernel_launch(void* const* d_in, const int* in_sizes, int n_in,
                              void* d_out, int out_size, void* d_ws, size_t ws_size,
                              hipStream_t stream) {
  const float* x   = (const float*)d_in[0];   // (B, 256, 8)
  const float* phi = (const float*)d_in[1];   // (6, 2, 18)
  const float* Wc  = (const float*)d_in[2];   // (6, 14)
  const float* bc  = (const float*)d_in[3];   // (6,)
  float* out = (float*)d_out;                 // y (B,256,6) | c (B,6) | h (B,6)
  const int B = in_sizes[0] / (TT * FD);

  float* Uf = (float*)d_ws;                   // 6 * 8192 floats = 192 KB

  build_unitaries<<<6, 64, 0, stream>>>(phi, Uf);
  qlstm_main<<<B / TB, 128, 0, stream>>>(x, Wc, bc, Uf, out, B);
  (void)n_in; (void)out_size; (void)ws_size;
}